// MultiHeadedAttention_11733850653130
// MI455X (gfx1250) — compile-verified
//
#include <hip/hip_runtime.h>
#include <hip/hip_bf16.h>

// ---------------------------------------------------------------------------
// MultiHeadedAttention with relative position embeddings (OpenNMT-style)
// B=4, S=1024, D=1024, H=16, DPH=64, MAX_REL=32
// All GEMMs run on v_wmma_f32_16x16x4_f32 (full f32 to match reference).
// Inner GEMM loops are software-pipelined (register double-buffering of
// 16-wide K chunks) so loads of chunk c+1 overlap the 16 WMMAs of chunk c.
// ---------------------------------------------------------------------------

#define B_    4
#define S_    1024
#define D_    1024
#define H_    16
#define DPH_  64
#define RELW_ 65     // 2*MAX_REL+1
#define RELP_ 80     // RELW_ padded to a multiple of 16

typedef float v2f __attribute__((ext_vector_type(2)));
typedef float v8f __attribute__((ext_vector_type(8)));

// WMMA f32 16x16x4 wrapper.
// A-fragment (16x4, MxK): lane L holds A[L%16, 2*(L/16)] and A[L%16, 2*(L/16)+1]
// B-fragment (4x16, KxN): lane L holds B[2*(L/16), L%16] and B[2*(L/16)+1, L%16]
// C/D (16x16): element r of lane L is D[r + 8*(L/16), L%16]
__device__ __forceinline__ v8f wmma_f32(v2f a, v2f b, v8f c) {
    return __builtin_amdgcn_wmma_f32_16x16x4_f32(
        /*neg_a=*/false, a, /*neg_b=*/false, b,
        /*c_mod=*/(short)0, c, /*reuse_a=*/false, /*reuse_b=*/false);
}

// Load one 16-K chunk (4 WMMA fragments) of A and NT B-tiles.
// ap / wp[] are already offset by +2*hl so fragment loads are plain v2f loads.
template <int NT>
__device__ __forceinline__ void load_chunk(const float* __restrict__ ap,
                                           const float* const* __restrict__ wp,
                                           int kk, v2f a[4], v2f b[NT][4])
{
#pragma unroll
    for (int u = 0; u < 4; ++u) {
        a[u] = *(const v2f*)(ap + kk + 4 * u);
#pragma unroll
        for (int nt = 0; nt < NT; ++nt)
            b[nt][u] = *(const v2f*)(wp[nt] + kk + 4 * u);
    }
}

template <int NT>
__device__ __forceinline__ void mac_chunk(v8f acc[NT], const v2f a[4], const v2f b[NT][4])
{
#pragma unroll
    for (int u = 0; u < 4; ++u)
#pragma unroll
        for (int nt = 0; nt < NT; ++nt)
            acc[nt] = wmma_f32(a[u], b[nt][u], acc[nt]);
}

// Software-pipelined K-loop: K must be a multiple of 32 (even # of 16-chunks).
template <int NT>
__device__ __forceinline__ void gemm_pipeline(const float* __restrict__ ap,
                                              const float* const* __restrict__ wp,
                                              int K, v8f acc[NT])
{
    v2f a0[4], a1[4];
    v2f b0[NT][4], b1[NT][4];
    const int chunks = K >> 4;
    load_chunk<NT>(ap, wp, 0, a0, b0);
    for (int c = 0; c + 2 <= chunks; c += 2) {
        load_chunk<NT>(ap, wp, (c + 1) << 4, a1, b1);   // prefetch odd chunk
        mac_chunk<NT>(acc, a0, b0);
        if (c + 2 < chunks)
            load_chunk<NT>(ap, wp, (c + 2) << 4, a0, b0); // prefetch even chunk
        mac_chunk<NT>(acc, a1, b1);
    }
}

// ---------------------------------------------------------------------------
// Kernel 1: out = scale * (X @ W^T)     X:[4096,1024] row-major, W:[N=1024,K=1024]
// mode 0: store head-split  [B,H,S,DPH]
// mode 1: store V transposed [B,H,DPH,S]
// mode 2: store plain        [M, N]
// One wave (32 threads) per 16x64 output strip. grid = 256 mtiles * 16 ngroups.
// ---------------------------------------------------------------------------
__global__ __launch_bounds__(32)
void gemm_xwT_kernel(const float* __restrict__ X, const float* __restrict__ W,
                     float* __restrict__ out, float scale, int mode)
{
    const int id   = blockIdx.x;
    const int mt   = id & 255;        // m-tile (16 rows)
    const int ng   = id >> 8;         // n-group (64 cols)
    const int lane = threadIdx.x;
    const int lr   = lane & 15;
    const int hl   = lane >> 4;

    const float* ap = X + (size_t)(mt * 16 + lr) * D_ + 2 * hl;
    const int nbase = ng * 64;
    const float* wp[4];
#pragma unroll
    for (int nt = 0; nt < 4; ++nt)
        wp[nt] = W + (size_t)(nbase + nt * 16 + lr) * D_ + 2 * hl;

    v8f acc[4] = {};
    gemm_pipeline<4>(ap, wp, D_, acc);

#pragma unroll
    for (int nt = 0; nt < 4; ++nt) {
#pragma unroll
        for (int r = 0; r < 8; ++r) {
            const int m  = mt * 16 + r + 8 * hl;
            const int n  = nbase + nt * 16 + lr;
            const float v = acc[nt][r] * scale;
            size_t idx;
            if (mode == 2) {
                idx = (size_t)m * D_ + n;
            } else {
                const int b = m >> 10, s = m & (S_ - 1);
                const int h = n >> 6,  d = n & (DPH_ - 1);
                if (mode == 0)
                    idx = ((size_t)((b * H_ + h) * S_ + s)) * DPH_ + d;   // [B,H,S,DPH]
                else
                    idx = ((size_t)((b * H_ + h) * DPH_ + d)) * S_ + s;   // [B,H,DPH,S]
            }
            out[idx] = v;
        }
    }
}

// ---------------------------------------------------------------------------
// Kernel 2: R = Qh @ rel_emb^T   Qh:[B*H*S, 64], rel_emb:[65,64] -> R:[B*H*S, 80]
// One wave per 16x80 strip. grid = (B*H*S)/16 = 4096. (~2% of flops, simple loop)
// ---------------------------------------------------------------------------
__global__ __launch_bounds__(32)
void relproj_kernel(const float* __restrict__ Qh, const float* __restrict__ rel,
                    float* __restrict__ R)
{
    const int mt   = blockIdx.x;
    const int lane = threadIdx.x;
    const int lr   = lane & 15;
    const int hl   = lane >> 4;

    const float* __restrict__ arow = Qh + (size_t)(mt * 16 + lr) * DPH_ + 2 * hl;

    v8f acc[5] = {};
    for (int kk = 0; kk < DPH_; kk += 4) {
        v2f a = *(const v2f*)(arow + kk);
#pragma unroll
        for (int nt = 0; nt < 5; ++nt) {
            const int t = nt * 16 + lr;
            v2f b;
            if (t < RELW_) b = *(const v2f*)(rel + (size_t)t * DPH_ + kk + 2 * hl);
            else           { b.x = 0.f; b.y = 0.f; }
            acc[nt] = wmma_f32(a, b, acc[nt]);
        }
    }

#pragma unroll
    for (int nt = 0; nt < 5; ++nt)
#pragma unroll
        for (int r = 0; r < 8; ++r)
            R[(size_t)(mt * 16 + r + 8 * hl) * RELP_ + nt * 16 + lr] = acc[nt][r];
}

// ---------------------------------------------------------------------------
// Kernel 3: scores[bh,q,k] = Qh[bh,q,:]·Kh[bh,k,:] + R[bh,q, clip(k-q)+32]
// One wave per 16(q) x 64(k) strip. grid = 64 bh * 64 qtiles * 16 kgroups.
// Writes pre-softmax scores into the attn output region.
// ---------------------------------------------------------------------------
__global__ __launch_bounds__(32)
void scores_kernel(const float* __restrict__ Qh, const float* __restrict__ Kh,
                   const float* __restrict__ R, float* __restrict__ scores)
{
    const int id   = blockIdx.x;
    const int kg   = id & 15;
    const int qt   = (id >> 4) & 63;
    const int bh   = id >> 10;
    const int lane = threadIdx.x;
    const int lr   = lane & 15;
    const int hl   = lane >> 4;

    const float* ap = Qh + ((size_t)bh * S_ + qt * 16 + lr) * DPH_ + 2 * hl;
    const float* wp[4];
#pragma unroll
    for (int nt = 0; nt < 4; ++nt)
        wp[nt] = Kh + ((size_t)bh * S_ + kg * 64 + nt * 16 + lr) * DPH_ + 2 * hl;

    v8f acc[4] = {};
    gemm_pipeline<4>(ap, wp, DPH_, acc);

#pragma unroll
    for (int nt = 0; nt < 4; ++nt) {
#pragma unroll
        for (int r = 0; r < 8; ++r) {
            const int q = qt * 16 + r + 8 * hl;
            const int k = kg * 64 + nt * 16 + lr;
            int ridx = k - q;
            ridx = (ridx < -32) ? -32 : (ridx > 32 ? 32 : ridx);
            ridx += 32;
            const float v = acc[nt][r] + R[((size_t)bh * S_ + q) * RELP_ + ridx];
            scores[((size_t)bh * S_ + q) * S_ + k] = v;
        }
    }
}

// ---------------------------------------------------------------------------
// Kernel 4: row softmax (with mask) + relative-position bucket sums.
// One 256-thread block per row (b,h,q). grid = B*H*S = 65536.
//   A65[row, 0]       = sum_{k <= q-32} attn
//   A65[row, 64]      = sum_{k >= q+32} attn
//   A65[row, t=1..63] = attn[q, q+t-32] (0 if out of range)
//   A65[row, 65..79]  = 0 (padding)
// ---------------------------------------------------------------------------
__global__ __launch_bounds__(256)
void softmax_kernel(float* __restrict__ attn, const unsigned char* __restrict__ mask,
                    float* __restrict__ A65)
{
    __shared__ float red[256];
    const int row = blockIdx.x;
    const int bh  = row >> 10;
    const int q   = row & (S_ - 1);
    const int b   = bh >> 4;
    const int tid = threadIdx.x;

    float* __restrict__ srow = attn + (size_t)row * S_;
    const unsigned char* __restrict__ mrow = mask + ((size_t)b * S_ + q) * S_;

    float vals[4];
    float mx = -3.0e38f;
#pragma unroll
    for (int i = 0; i < 4; ++i) {
        const int k = tid + i * 256;
        float v = srow[k];
        if (mrow[k]) v = -1.0e18f;
        vals[i] = v;
        mx = fmaxf(mx, v);
    }
    red[tid] = mx; __syncthreads();
    for (int off = 128; off > 0; off >>= 1) {
        if (tid < off) red[tid] = fmaxf(red[tid], red[tid + off]);
        __syncthreads();
    }
    mx = red[0]; __syncthreads();

    float sum = 0.f;
#pragma unroll
    for (int i = 0; i < 4; ++i) { vals[i] = __expf(vals[i] - mx); sum += vals[i]; }
    red[tid] = sum; __syncthreads();
    for (int off = 128; off > 0; off >>= 1) {
        if (tid < off) red[tid] += red[tid + off];
        __syncthreads();
    }
    const float inv = 1.0f / red[0]; __syncthreads();

    float s0 = 0.f, s64 = 0.f;
#pragma unroll
    for (int i = 0; i < 4; ++i) {
        const int k = tid + i * 256;
        const float a = vals[i] * inv;
        srow[k] = a;
        if (k <= q - 32) s0  += a;
        if (k >= q + 32) s64 += a;
    }
    red[tid] = s0; __syncthreads();
    for (int off = 128; off > 0; off >>= 1) {
        if (tid < off) red[tid] += red[tid + off];
        __syncthreads();
    }
    s0 = red[0]; __syncthreads();
    red[tid] = s64; __syncthreads();
    for (int off = 128; off > 0; off >>= 1) {
        if (tid < off) red[tid] += red[tid + off];
        __syncthreads();
    }
    s64 = red[0]; __syncthreads();   // also guarantees srow[] writes are visible

    float* __restrict__ arow = A65 + (size_t)row * RELP_;
    if (tid == 0) { arow[0] = s0; arow[64] = s64; }
    if (tid >= 1 && tid <= 63) {
        const int k = q + tid - 32;
        arow[tid] = (k >= 0 && k < S_) ? srow[k] : 0.f;
    }
    if (tid >= 65 && tid < RELP_) arow[tid] = 0.f;
}

// ---------------------------------------------------------------------------
// Kernel 5: ctx[bh,q,:] = attn[bh,q,:] @ V  +  A65[bh,q,:] @ rel_emb
// Vt is [B,H,DPH,S] so B-fragment loads are consecutive along k.
// One wave per 16(q) x 64(d) tile. grid = 64 bh * 64 qtiles.
// Stores ctx directly in unshaped [B,S,D] layout for the output projection.
// ---------------------------------------------------------------------------
__global__ __launch_bounds__(32)
void ctx_kernel(const float* __restrict__ attn, const float* __restrict__ Vt,
                const float* __restrict__ A65, const float* __restrict__ rel,
                float* __restrict__ ctx)
{
    const int id   = blockIdx.x;
    const int qt   = id & 63;
    const int bh   = id >> 6;
    const int lane = threadIdx.x;
    const int lr   = lane & 15;
    const int hl   = lane >> 4;

    const float* ap = attn + ((size_t)bh * S_ + qt * 16 + lr) * S_ + 2 * hl;
    const float* wp[4];
#pragma unroll
    for (int nt = 0; nt < 4; ++nt)
        wp[nt] = Vt + ((size_t)bh * DPH_ + nt * 16 + lr) * S_ + 2 * hl;

    v8f acc[4] = {};
    gemm_pipeline<4>(ap, wp, S_, acc);

    // relative-value contribution: A65(row, 0..79) @ rel_emb(65x64), zero-padded
    const float* __restrict__ arow2 = A65 + ((size_t)bh * S_ + qt * 16 + lr) * RELP_ + 2 * hl;
    for (int kk = 0; kk < RELP_; kk += 4) {
        v2f a = *(const v2f*)(arow2 + kk);
        const int t0 = kk + 2 * hl;
#pragma unroll
        for (int nt = 0; nt < 4; ++nt) {
            const int n = nt * 16 + lr;
            v2f b;
            b.x = (t0     < RELW_) ? rel[(size_t)t0       * DPH_ + n] : 0.f;
            b.y = (t0 + 1 < RELW_) ? rel[(size_t)(t0 + 1) * DPH_ + n] : 0.f;
            acc[nt] = wmma_f32(a, b, acc[nt]);
        }
    }

    const int bb = bh >> 4;
    const int hh = bh & 15;
#pragma unroll
    for (int nt = 0; nt < 4; ++nt) {
#pragma unroll
        for (int r = 0; r < 8; ++r) {
            const int q = qt * 16 + r + 8 * hl;
            const int d = nt * 16 + lr;
            ctx[((size_t)(bb * S_ + q)) * D_ + hh * DPH_ + d] = acc[nt][r];
        }
    }
}

// ---------------------------------------------------------------------------
// Launch
// ---------------------------------------------------------------------------
extern "C" void kernel_launch(void* const* d_in, const int* in_sizes, int n_in,
                              void* d_out, int out_size, void* d_ws, size_t ws_size,
                              hipStream_t stream)
{
    const float*         key_in   = (const float*)d_in[0];
    const float*         value_in = (const float*)d_in[1];
    const float*         query_in = (const float*)d_in[2];
    const unsigned char* mask     = (const unsigned char*)d_in[3];
    const float*         w_q      = (const float*)d_in[4];
    const float*         w_k      = (const float*)d_in[5];
    const float*         w_v      = (const float*)d_in[6];
    const float*         w_o      = (const float*)d_in[7];
    const float*         rel_emb  = (const float*)d_in[8];

    float* out  = (float*)d_out;                       // [B,S,D] output
    float* attn = out + (size_t)B_ * S_ * D_;          // [B,H,S,S] attn output

    // workspace layout (floats)
    float* ws   = (float*)d_ws;
    const size_t nProj = (size_t)B_ * H_ * S_ * DPH_;  // 4,194,304
    const size_t nRel  = (size_t)B_ * H_ * S_ * RELP_; // 5,242,880
    float* Qh   = ws;
    float* Kh   = Qh + nProj;
    float* Vt   = Kh + nProj;
    float* Rbuf = Vt + nProj;
    float* A65  = Rbuf + nRel;
    float* ctx  = A65 + nRel;

    const dim3 wblk(32);

    // Projections (Q scaled by 1/sqrt(DPH) = 1/8)
    gemm_xwT_kernel<<<4096, wblk, 0, stream>>>(query_in, w_q, Qh, 0.125f, 0);
    gemm_xwT_kernel<<<4096, wblk, 0, stream>>>(key_in,   w_k, Kh, 1.0f,   0);
    gemm_xwT_kernel<<<4096, wblk, 0, stream>>>(value_in, w_v, Vt, 1.0f,   1);

    // R = Qh @ rel_emb^T  (65 valid cols, padded to 80)
    relproj_kernel<<<4096, wblk, 0, stream>>>(Qh, rel_emb, Rbuf);

    // scores = Q K^T + gather(R)  -> written into attn region
    scores_kernel<<<64 * 64 * 16, wblk, 0, stream>>>(Qh, Kh, Rbuf, attn);

    // softmax + bucket sums
    softmax_kernel<<<B_ * H_ * S_, 256, 0, stream>>>(attn, mask, A65);

    // ctx = attn @ V + A65 @ rel_emb  (stored unshaped [B,S,D])
    ctx_kernel<<<64 * 64, wblk, 0, stream>>>(attn, Vt, A65, rel_emb, ctx);

    // output = ctx @ w_o^T
    gemm_xwT_kernel<<<4096, wblk, 0, stream>>>(ctx, w_o, out, 1.0f, 2);
}